// SAGPool_27565100105878
// MI455X (gfx1250) — compile-verified
//
#include <hip/hip_runtime.h>
#include <math.h>

// Problem constants (match reference)
#define N_GRAPHS 64
#define NPG 1024
#define NN (N_GRAPHS * NPG)   // 65536 nodes
#define DD 128                // embed dim
#define DEG 16
#define EE (NN * DEG)         // 1048576 edges
#define DEA 8                 // edge attr dim
#define KK 820                // kept per graph
#define NK (N_GRAPHS * KK)    // 52480 kept nodes total

typedef __attribute__((ext_vector_type(2))) float v2f;
typedef __attribute__((ext_vector_type(8))) float v8f;

// ---------------------------------------------------------------------------
// Kernel 1: per-node dots via V_WMMA_F32_16X16X4_F32 (branch-free inner loop).
//   A (16x4 f32): lane L -> row M=L&15, K half = (L>>4)*2 .. +1  (2 VGPRs)
//   B (4x16 f32): lane L -> col N=L&15, rows k0+(L>>4)*2 and +1  (2 VGPRs)
//   C/D (16x16 f32, 8 VGPRs): lane L -> col N=L&15, rows v + 8*(L>>4)
// B column 0 = W_rel, column 1 = W_root, others zeroed via a lane mask
// (lanes with col>=2 load valid Wroot data and multiply by 0 -> no EXEC
//  divergence around the WMMA, which requires EXEC == all ones).
// Writes p_rel[node] and attn[node] = b_rel + p_root[node].
// ---------------------------------------------------------------------------
__global__ __launch_bounds__(128) void dot_wmma_kernel(
    const float* __restrict__ x, const float* __restrict__ Wrel,
    const float* __restrict__ Wroot, const float* __restrict__ brel,
    float* __restrict__ p_rel, float* __restrict__ attn) {
  const int lane = threadIdx.x & 31;
  const int wave = threadIdx.x >> 5;
  const int base = (blockIdx.x * 4 + wave) * 16;  // 16 nodes per wave
  const int M = lane & 15;                        // A row
  const int n = lane & 15;                        // output column
  const int koff = (lane >> 4) * 2;               // which half of K=4 slice

  const float* xrow = x + (size_t)(base + M) * DD + koff;
  const float* bptr = ((n == 0) ? Wrel : Wroot) + koff;  // uniform per lane
  const float bmask = (n <= 1) ? 1.0f : 0.0f;

  v8f c = {0.f, 0.f, 0.f, 0.f, 0.f, 0.f, 0.f, 0.f};
#pragma unroll 8
  for (int k0 = 0; k0 < DD; k0 += 4) {
    v2f a = *(const v2f*)(xrow + k0);
    v2f b = *(const v2f*)(bptr + k0);
    b.x *= bmask;
    b.y *= bmask;
    c = __builtin_amdgcn_wmma_f32_16x16x4_f32(false, a, false, b, (short)0, c,
                                              false, false);
  }

  // Extract column 0 (p_rel) from lanes 0/16, column 1 (p_root) from 1/17.
  if (n <= 1) {
    const int rbase = (lane >> 4) * 8;
    const float bb = brel[0];
#pragma unroll
    for (int v = 0; v < 8; ++v) {
      const float val = c[v];
      const int node = base + rbase + v;
      if (n == 0)
        p_rel[node] = val;
      else
        attn[node] = bb + val;
    }
  }
}

// ---------------------------------------------------------------------------
// Kernel 2: attn[dst] += p_rel[src] over all edges (scalar, 4B/edge payload).
// ---------------------------------------------------------------------------
__global__ __launch_bounds__(256) void edge_agg_kernel(
    const int* __restrict__ src, const int* __restrict__ dst,
    const float* __restrict__ p_rel, float* __restrict__ attn) {
  const int e = blockIdx.x * 256 + threadIdx.x;
  if (e < EE) atomicAdd(&attn[dst[e]], p_rel[src[e]]);
}

// ---------------------------------------------------------------------------
// Kernel 3: per-graph top-K via LDS bitonic sort on (score desc, idx asc).
// One 1024-thread block per graph. Also builds cluster[] relabeling.
// ---------------------------------------------------------------------------
__global__ __launch_bounds__(1024) void topk_kernel(
    const float* __restrict__ attn, const float* __restrict__ selw,
    const int* __restrict__ gi, int* __restrict__ perm,
    float* __restrict__ weight, int* __restrict__ cluster,
    float* __restrict__ out_batch) {
  __shared__ float sk[NPG];
  __shared__ int si[NPG];
  const int g = blockIdx.x;
  const int t = threadIdx.x;
  const float sgn = (selw[0] >= 0.f) ? 1.f : -1.f;  // w/|w|

  sk[t] = tanhf(attn[g * NPG + t] * sgn);
  si[t] = t;
  __syncthreads();

  for (int k = 2; k <= NPG; k <<= 1) {
    for (int j = k >> 1; j > 0; j >>= 1) {
      const int l = t ^ j;
      if (l > t) {
        const float ki = sk[t], kl = sk[l];
        const int ii = si[t], il = si[l];
        const bool before = (ki > kl) || (ki == kl && ii < il);
        const bool up = ((t & k) == 0);
        if (up ? !before : before) {
          sk[t] = kl; sk[l] = ki;
          si[t] = il; si[l] = ii;
        }
      }
      __syncthreads();
    }
  }

  const int node = g * NPG + si[t];
  if (t < KK) {
    const int r = g * KK + t;
    perm[r] = node;
    weight[r] = sk[t];
    cluster[node] = r;
    out_batch[r] = (float)gi[node];
  } else {
    cluster[node] = -1;  // si[] is a permutation -> every node covered
  }
}

// ---------------------------------------------------------------------------
// Kernel 4: new_input_feature[row] = x[perm[row]] * weight[row]  (float4)
// ---------------------------------------------------------------------------
__global__ __launch_bounds__(256) void feat_kernel(
    const float* __restrict__ x, const int* __restrict__ perm,
    const float* __restrict__ weight, float* __restrict__ out0) {
  const int row = blockIdx.x * 8 + (threadIdx.x >> 5);
  const int lane = threadIdx.x & 31;  // 32 float4 per 128-float row
  if (row < NK) {
    const int node = perm[row];
    const float w = weight[row];
    const float4* xr = (const float4*)(x + (size_t)node * DD);
    float4* o = (float4*)(out0 + (size_t)row * DD);
    float4 v = xr[lane];
    v.x *= w; v.y *= w; v.z *= w; v.w *= w;
    o[lane] = v;
  }
}

// ---------------------------------------------------------------------------
// Kernel 5: fill new_edge_index with -1 and new_edge_attr with 0 (float4).
// ---------------------------------------------------------------------------
__global__ __launch_bounds__(256) void init_edges_kernel(
    float* __restrict__ out_idx, float* __restrict__ out_attr) {
  const int i = blockIdx.x * 256 + threadIdx.x;
  const int n1 = (2 * EE) / 4;        // float4 count for edge_index
  const int n3 = (EE * DEA) / 4;      // float4 count for edge_attr
  if (i < n1) ((float4*)out_idx)[i] = make_float4(-1.f, -1.f, -1.f, -1.f);
  if (i < n3) ((float4*)out_attr)[i] = make_float4(0.f, 0.f, 0.f, 0.f);
}

// ---------------------------------------------------------------------------
// Kernel 6: per-block valid-edge counts (4 contiguous edges per thread).
// ---------------------------------------------------------------------------
__global__ __launch_bounds__(256) void edge_count_kernel(
    const int* __restrict__ src, const int* __restrict__ dst,
    const int* __restrict__ cluster, unsigned* __restrict__ blockSums) {
  __shared__ unsigned red[256];
  const int t = threadIdx.x;
  const int base = (blockIdx.x * 256 + t) * 4;
  unsigned cnt = 0;
#pragma unroll
  for (int q = 0; q < 4; ++q) {
    const int e = base + q;
    cnt += (cluster[src[e]] >= 0 && cluster[dst[e]] >= 0) ? 1u : 0u;
  }
  red[t] = cnt;
  __syncthreads();
  for (int off = 128; off > 0; off >>= 1) {
    if (t < off) red[t] += red[t + off];
    __syncthreads();
  }
  if (t == 0) blockSums[blockIdx.x] = red[0];
}

// ---------------------------------------------------------------------------
// Kernel 7: exclusive scan of the 1024 block sums (single 1024-thread block).
// ---------------------------------------------------------------------------
__global__ __launch_bounds__(1024) void scan_blocks_kernel(
    const unsigned* __restrict__ blockSums, unsigned* __restrict__ blockOffs) {
  __shared__ unsigned tmp[1024];
  const int t = threadIdx.x;
  const unsigned own = blockSums[t];
  tmp[t] = own;
  __syncthreads();
  for (int off = 1; off < 1024; off <<= 1) {
    const unsigned v = (t >= off) ? tmp[t - off] : 0u;
    __syncthreads();
    tmp[t] += v;
    __syncthreads();
  }
  blockOffs[t] = tmp[t] - own;  // exclusive
}

// ---------------------------------------------------------------------------
// Kernel 8: stable scatter of valid edges to the compacted front.
// ---------------------------------------------------------------------------
__global__ __launch_bounds__(256) void edge_scatter_kernel(
    const int* __restrict__ src, const int* __restrict__ dst,
    const int* __restrict__ cluster, const float* __restrict__ eattr,
    const unsigned* __restrict__ blockOffs, float* __restrict__ out_idx,
    float* __restrict__ out_attr) {
  __shared__ unsigned tcnt[256];
  const int t = threadIdx.x;
  const int base = (blockIdx.x * 256 + t) * 4;
  int rr[4], cc[4];
  unsigned m[4];
  unsigned cnt = 0;
#pragma unroll
  for (int q = 0; q < 4; ++q) {
    const int e = base + q;
    rr[q] = cluster[src[e]];
    cc[q] = cluster[dst[e]];
    m[q] = (rr[q] >= 0 && cc[q] >= 0) ? 1u : 0u;
    cnt += m[q];
  }
  tcnt[t] = cnt;
  __syncthreads();
  for (int off = 1; off < 256; off <<= 1) {
    const unsigned v = (t >= off) ? tcnt[t - off] : 0u;
    __syncthreads();
    tcnt[t] += v;
    __syncthreads();
  }
  unsigned pos = blockOffs[blockIdx.x] + tcnt[t] - cnt;  // stable position
#pragma unroll
  for (int q = 0; q < 4; ++q) {
    if (m[q]) {
      const int e = base + q;
      out_idx[pos] = (float)rr[q];
      out_idx[(size_t)EE + pos] = (float)cc[q];
      const float4* ar = (const float4*)(eattr + (size_t)e * DEA);
      float4* ow = (float4*)(out_attr + (size_t)pos * DEA);
      ow[0] = ar[0];
      ow[1] = ar[1];
      ++pos;
    }
  }
}

// ---------------------------------------------------------------------------
extern "C" void kernel_launch(void* const* d_in, const int* in_sizes, int n_in,
                              void* d_out, int out_size, void* d_ws,
                              size_t ws_size, hipStream_t stream) {
  (void)in_sizes; (void)n_in; (void)out_size; (void)ws_size;

  const float* x     = (const float*)d_in[0];
  const int*   adj   = (const int*)d_in[1];
  const int*   gi    = (const int*)d_in[2];
  const float* eattr = (const float*)d_in[3];
  const float* Wrel  = (const float*)d_in[4];
  const float* brel  = (const float*)d_in[5];
  const float* Wroot = (const float*)d_in[6];
  const float* selw  = (const float*)d_in[7];
  const int* src = adj;
  const int* dst = adj + EE;

  // Outputs concatenated flat in return order (as float).
  float* out0 = (float*)d_out;                  // new_input_feature [NK, D]
  float* out1 = out0 + (size_t)NK * DD;         // new_edge_index   [2, E]
  float* out2 = out1 + (size_t)2 * EE;          // new_batch        [NK]
  float* out3 = out2 + (size_t)NK;              // new_edge_attr    [E, DEA]

  // Workspace layout (~1.2 MB).
  char* ws = (char*)d_ws;
  float* p_rel = (float*)ws;            ws += (size_t)NN * 4;
  float* attn  = (float*)ws;            ws += (size_t)NN * 4;
  int* cluster = (int*)ws;              ws += (size_t)NN * 4;
  int* perm    = (int*)ws;              ws += (size_t)NK * 4;
  float* weight = (float*)ws;           ws += (size_t)NK * 4;
  unsigned* blockSums = (unsigned*)ws;  ws += 1024 * 4;
  unsigned* blockOffs = (unsigned*)ws;  ws += 1024 * 4;

  dot_wmma_kernel<<<NN / 64, 128, 0, stream>>>(x, Wrel, Wroot, brel, p_rel,
                                               attn);
  edge_agg_kernel<<<EE / 256, 256, 0, stream>>>(src, dst, p_rel, attn);
  topk_kernel<<<N_GRAPHS, NPG, 0, stream>>>(attn, selw, gi, perm, weight,
                                            cluster, out2);
  feat_kernel<<<(NK + 7) / 8, 256, 0, stream>>>(x, perm, weight, out0);
  init_edges_kernel<<<(EE * DEA / 4) / 256, 256, 0, stream>>>(out1, out3);
  edge_count_kernel<<<EE / 1024, 256, 0, stream>>>(src, dst, cluster,
                                                   blockSums);
  scan_blocks_kernel<<<1, 1024, 0, stream>>>(blockSums, blockOffs);
  edge_scatter_kernel<<<EE / 1024, 256, 0, stream>>>(src, dst, cluster, eattr,
                                                     blockOffs, out1, out3);
}